// MambaDecoderLayer_31516470018165
// MI455X (gfx1250) — compile-verified
//
#include <hip/hip_runtime.h>
#include <math.h>

// ---------------- problem constants (match reference) ----------------
#define D_MODEL   1024
#define D_INNER   2048
#define D_XB       512
#define D_STATE     16
#define D_CONV       4
#define DT_RANK     64
#define NXB         32                       // D_XB / D_STATE
#define H_HEADS    128                       // D_INNER / D_STATE
#define REP          4                       // D_INNER / D_XB
#define PCH         16                       // channels per head
#define BATCH        2
#define SEQ       2048
#define E_DIM     (2*D_INNER + 2*D_XB + DT_RANK)   // 5184
#define M_ROWS    (BATCH*SEQ)                      // 4096

// zxbcdt segment offsets
#define OFF_Z   0
#define OFF_X   (D_INNER)                 // 2048
#define OFF_B   (D_INNER + D_XB)          // 2560
#define OFF_C   (D_INNER + 2*D_XB)        // 3072
#define OFF_DT  (2*D_INNER + 2*D_XB)      // 5120

typedef __attribute__((ext_vector_type(2))) float v2f;
typedef __attribute__((ext_vector_type(8))) float v8f;

// D = A(16x4, f32) * B(4x16, f32) + C(16x16, f32)  — CDNA5 fp32 WMMA path.
// 8-arg pattern: (neg_a, A, neg_b, B, c_mod, C, reuse_a, reuse_b)
__device__ __forceinline__ v8f wmma_f32_16x16x4(v2f a, v2f b, v8f c) {
  return __builtin_amdgcn_wmma_f32_16x16x4_f32(
      /*neg_a=*/false, a, /*neg_b=*/false, b,
      /*c_mod=*/(short)0, c, /*reuse_a=*/false, /*reuse_b=*/false);
}

// --------------------------------------------------------------------
// fp32 WMMA GEMM with 4x4 register fragment blocking:
//   C[M,N] = A[M,K] (row-major, lda) * W[N,K]^T (ldb)
// Each wave computes a 64x64 output tile = 4x4 grid of 16x16 WMMA
// tiles.  Per K-step (K+=4): 8 aligned 8B loads -> 16 WMMAs on 16
// independent accumulators (no back-to-back RAW on the matrix pipe),
// i.e. 16 flops per loaded byte, so neither VMEM issue rate nor L2
// bandwidth caps the fp32 matrix pipe.  Register budget ~180 VGPRs
// (16 x v8f acc + 8 x v2f frags + addresses) -- no spills.
// A-operand layout (ISA 7.12.2, 32-bit A 16x4): lane l holds row
// M=l%16, K = 2*(l/16)+{0,1}.  B mirrors the half-wave K split with
// N=l%16.  C/D: VGPR v, lanes 0-15 -> M=v, lanes 16-31 -> M=v+8.
// EPI: 0 = none, 1 = softplus(x + bias[n])  (dt projection)
// --------------------------------------------------------------------
template<int EPI>
__global__ void __launch_bounds__(256)
wmma_gemm_f32(const float* __restrict__ A, int lda,
              const float* __restrict__ W, int ldb,
              float* __restrict__ C, int ldc,
              int M, int N, int K,
              const float* __restrict__ bias)
{
  const int lane = threadIdx.x & 31;
  const int wave = threadIdx.x >> 5;
  const int tilesN = N >> 6;                 // 64-wide supertiles
  const int totalTiles = (M >> 6) * tilesN;
  const int tile = blockIdx.x * (blockDim.x >> 5) + wave;
  if (tile >= totalTiles) return;

  const int tm = tile / tilesN;
  const int tn = tile - tm * tilesN;

  const int r    = lane & 15;        // row-in-frag (A) / col-in-frag (B)
  const int klo  = (lane >> 4) << 1; // half-wave K offset {0, 2}

  const float* __restrict__ Ar[4];
  const float* __restrict__ Br[4];
#pragma unroll
  for (int i = 0; i < 4; ++i) {
    Ar[i] = A + (size_t)(tm * 64 + i * 16 + r) * lda + klo;
    Br[i] = W + (size_t)(tn * 64 + i * 16 + r) * ldb + klo;
  }

  v8f acc[4][4];
#pragma unroll
  for (int i = 0; i < 4; ++i)
#pragma unroll
    for (int j = 0; j < 4; ++j) acc[i][j] = {};

  for (int k = 0; k < K; k += 4) {
    v2f a[4], b[4];
#pragma unroll
    for (int i = 0; i < 4; ++i) a[i] = *(const v2f*)(Ar[i] + k);  // 8B aligned
#pragma unroll
    for (int j = 0; j < 4; ++j) b[j] = *(const v2f*)(Br[j] + k);
#pragma unroll
    for (int i = 0; i < 4; ++i)
#pragma unroll
      for (int j = 0; j < 4; ++j)
        acc[i][j] = wmma_f32_16x16x4(a[i], b[j], acc[i][j]);
  }

  const int half = lane >> 4;
#pragma unroll
  for (int i = 0; i < 4; ++i) {
#pragma unroll
    for (int j = 0; j < 4; ++j) {
      const int cn = tn * 64 + j * 16 + r;
#pragma unroll
      for (int v = 0; v < 8; ++v) {
        const int cm = tm * 64 + i * 16 + v + (half << 3);
        float val = acc[i][j][v];
        if (EPI == 1) {
          val += bias[cn];
          val = (val > 20.f) ? val : log1pf(__expf(val));  // softplus
        }
        C[(size_t)cm * ldc + cn] = val;
      }
    }
  }
}

// --------------------------------------------------------------------
// repeat_kv(x) + depthwise causal conv1d + SiLU.
// One thread per (b, l, d).  Source xb channel: (d/64)*16 + d%16.
// --------------------------------------------------------------------
__global__ void __launch_bounds__(256)
conv_silu_kernel(const float* __restrict__ zx,
                 const float* __restrict__ conv_w,
                 const float* __restrict__ conv_b,
                 float* __restrict__ xconv)
{
  int idx = blockIdx.x * blockDim.x + threadIdx.x;
  if (idx >= BATCH * SEQ * D_INNER) return;
  const int d = idx % D_INNER;
  const int l = (idx / D_INNER) % SEQ;
  const int b = idx / (D_INNER * SEQ);
  const int src = ((d >> 6) << 4) + (d & 15);   // repeat_kv mapping

  const float* __restrict__ w = conv_w + d * D_CONV;
  float acc = conv_b[d];
#pragma unroll
  for (int t = 0; t < D_CONV; ++t) {
    const int ll = l - (D_CONV - 1) + t;        // causal left-pad
    if (ll >= 0) {
      float xv = zx[(size_t)(b * SEQ + ll) * E_DIM + OFF_X + src];
      acc = fmaf(w[t], xv, acc);
    }
  }
  acc = acc / (1.f + __expf(-acc));             // SiLU
  xconv[(size_t)idx] = acc;
}

// --------------------------------------------------------------------
// Selective scan. One block per (b, head); 256 threads = one per state
// element (p = tid/16, n = tid%16).  Recurrence critical path is a
// single FMA per step; dA = exp(dt*A) is off the dependency chain.
// y[p] = sum_n h[p][n]*C[n] via 16-lane xor-shuffle reduction.
// Gate (SiLU(z)) and D-skip fused into the store.  All per-step
// operands stream from buffers that fit in the 192MB L2.
// --------------------------------------------------------------------
__global__ void __launch_bounds__(256)
scan_kernel(const float* __restrict__ zx,
            const float* __restrict__ xconv,
            const float* __restrict__ dtbuf,
            const float* __restrict__ Amat,
            const float* __restrict__ Dvec,
            float* __restrict__ Y)
{
  const int b = blockIdx.x / H_HEADS;
  const int h = blockIdx.x % H_HEADS;
  const int p = threadIdx.x >> 4;
  const int n = threadIdx.x & 15;

  const float An = Amat[(size_t)(h * PCH + p) * D_STATE + n];  // = -(n+1)
  const float Dv = Dvec[h * PCH + p];

  const size_t chan = (size_t)h * PCH + p;           // inner channel
  const size_t boff = OFF_B + (size_t)(h / REP) * D_STATE + n;
  const size_t coff = OFF_C + (size_t)h * D_STATE + n;
  const size_t zoff = OFF_Z + chan;

  float hstate = 0.f;
  size_t row = (size_t)b * SEQ;
  for (int l = 0; l < SEQ; ++l, ++row) {
    const float dt = dtbuf[row * D_INNER + chan];
    const float xv = xconv[row * D_INNER + chan];
    const float Bv = zx[row * E_DIM + boff];
    const float Cv = zx[row * E_DIM + coff];

    const float a = __expf(dt * An);
    hstate = fmaf(a, hstate, dt * xv * Bv);

    float yc = hstate * Cv;
    yc += __shfl_xor(yc, 1, 32);
    yc += __shfl_xor(yc, 2, 32);
    yc += __shfl_xor(yc, 4, 32);
    yc += __shfl_xor(yc, 8, 32);

    if (n == 0) {
      const float zval = zx[row * E_DIM + zoff];
      const float gate = zval / (1.f + __expf(-zval));
      Y[row * D_INNER + chan] = (yc + Dv * xv) * gate;
    }
  }
}

// --------------------------------------------------------------------
extern "C" void kernel_launch(void* const* d_in, const int* in_sizes, int n_in,
                              void* d_out, int out_size, void* d_ws, size_t ws_size,
                              hipStream_t stream) {
  const float* hidden     = (const float*)d_in[0];
  const float* in_proj_w  = (const float*)d_in[1];
  const float* conv_w     = (const float*)d_in[2];
  const float* conv_b     = (const float*)d_in[3];
  const float* dt_proj_w  = (const float*)d_in[4];
  const float* dt_proj_b  = (const float*)d_in[5];
  const float* Amat       = (const float*)d_in[6];
  const float* Dvec       = (const float*)d_in[7];
  const float* out_proj_w = (const float*)d_in[8];
  float* out = (float*)d_out;

  // workspace layout (all fully overwritten each call -> deterministic)
  float* zx    = (float*)d_ws;                                   // M x 5184
  float* xconv = zx    + (size_t)M_ROWS * E_DIM;                 // M x 2048
  float* dtbuf = xconv + (size_t)M_ROWS * D_INNER;               // M x 2048
  float* Ybuf  = dtbuf + (size_t)M_ROWS * D_INNER;               // M x 2048

  // 1) in_proj: zxbcdt = hidden @ in_proj_w^T   (M=4096, N=5184, K=1024)
  {
    int tiles = (M_ROWS / 64) * (E_DIM / 64);
    wmma_gemm_f32<0><<<(tiles + 7) / 8, 256, 0, stream>>>(
        hidden, D_MODEL, in_proj_w, D_MODEL, zx, E_DIM,
        M_ROWS, E_DIM, D_MODEL, nullptr);
  }

  // 2) repeat_kv + causal depthwise conv1d + SiLU
  {
    int n = BATCH * SEQ * D_INNER;
    conv_silu_kernel<<<(n + 255) / 256, 256, 0, stream>>>(zx, conv_w, conv_b, xconv);
  }

  // 3) dt = softplus(dt_in @ dt_proj_w^T + bias)  (M=4096, N=2048, K=64)
  {
    int tiles = (M_ROWS / 64) * (D_INNER / 64);
    wmma_gemm_f32<1><<<(tiles + 7) / 8, 256, 0, stream>>>(
        zx + OFF_DT, E_DIM, dt_proj_w, DT_RANK, dtbuf, D_INNER,
        M_ROWS, D_INNER, DT_RANK, dt_proj_b);
  }

  // 4) selective scan + D-skip + SiLU(z) gate
  scan_kernel<<<BATCH * H_HEADS, 256, 0, stream>>>(zx, xconv, dtbuf, Amat, Dvec, Ybuf);

  // 5) out = Y @ out_proj_w^T   (M=4096, N=1024, K=2048)
  {
    int tiles = (M_ROWS / 64) * (D_MODEL / 64);
    wmma_gemm_f32<0><<<(tiles + 7) / 8, 256, 0, stream>>>(
        Ybuf, D_INNER, out_proj_w, D_INNER, out, D_MODEL,
        M_ROWS, D_MODEL, D_INNER, nullptr);
  }
}